// GCN_66297115181508
// MI455X (gfx1250) — compile-verified
//
#include <hip/hip_runtime.h>

#define HIDDEN   128
#define NGRAPHS  512
#define NCLASSES 10

typedef __attribute__((ext_vector_type(2))) float v2f;
typedef __attribute__((ext_vector_type(8))) float v8f;
typedef __attribute__((address_space(3))) float lds_f;

// ---------------- normalization ----------------
__global__ void k_deg_init(float* deg, int n) {
  int i = blockIdx.x * blockDim.x + threadIdx.x;
  if (i < n) deg[i] = 1.0f;                    // self-loop contributes 1
}

__global__ void k_deg_accum(const int* __restrict__ dst, float* deg, int e) {
  int i = blockIdx.x * blockDim.x + threadIdx.x;
  if (i < e) atomicAdd(&deg[dst[i]], 1.0f);
}

__global__ void k_deg_to_dis(float* deg, int n) {
  int i = blockIdx.x * blockDim.x + threadIdx.x;
  if (i < n) deg[i] = rsqrtf(deg[i]);          // deg >= 1 always (self-loop)
}

__global__ void k_edge_norm(const int* __restrict__ src, const int* __restrict__ dst,
                            const float* __restrict__ dis, float* __restrict__ norm, int e) {
  int i = blockIdx.x * blockDim.x + threadIdx.x;
  if (i < e) norm[i] = dis[src[i]] * dis[dst[i]];
}

// ---------------- layer 1 transform (K = NUM_FEAT = 1) ----------------
__global__ void k_l1_transform(const float* __restrict__ x, const float* __restrict__ W1,
                               float* __restrict__ t, int n) {
  int idx = blockIdx.x * blockDim.x + threadIdx.x;     // one float4 per thread
  int total = n * (HIDDEN / 4);
  if (idx >= total) return;
  int node = idx >> 5;
  int c4 = (idx & 31) << 2;
  float xv = x[node];
  float4 wv = *(const float4*)(W1 + c4);
  float4 r = make_float4(xv * wv.x, xv * wv.y, xv * wv.z, xv * wv.w);
  *(float4*)(t + (size_t)node * HIDDEN + c4) = r;
}

// ---------------- dense transform, layers 2..5: [n,128] @ [128,128] ----------------
// Block = 8 wave32s; each wave owns one 16x16 f32 tile (V_WMMA_F32_16X16X4_F32).
// The shared 16x128 A stripe is staged into LDS once per block with
// GLOBAL_LOAD_ASYNC_TO_LDS_B128 (row stride padded to 132 floats so the
// 16-row, stride-528B ds_load_b64 fragment reads are bank-conflict-free).
__global__ void k_mm128_wmma(const float* __restrict__ H, const float* __restrict__ W,
                             float* __restrict__ T, int n) {
  constexpr int APAD = 132;                    // padded LDS row stride (floats)
  __shared__ __align__(16) float ldsA[16 * APAD];

  int tid  = threadIdx.x;
  int wave = tid >> 5;
  int lane = tid & 31;
  int half = lane >> 4;                        // 0: lanes 0-15, 1: lanes 16-31
  int lm   = lane & 15;
  int row0 = blockIdx.x * 16;

  // ---- async-stage A stripe (16 rows x 128 floats = 8KB): 2 x B128 per thread
  #pragma unroll
  for (int c = tid; c < 512; c += 256) {
    int r    = c >> 5;                         // 32 16-byte chunks per row
    int col4 = (c & 31) << 2;
    int arow = row0 + r;
    if (arow >= n) arow = n - 1;               // clamp tail loads
    const float* gp = H + (size_t)arow * HIDDEN + col4;
    lds_f* lp = (lds_f*)&ldsA[r * APAD + col4];
    asm volatile("global_load_async_to_lds_b128 %0, %1, off"
                 :: "v"(lp), "v"(gp) : "memory");
  }
  asm volatile("s_wait_asynccnt 0x0" ::: "memory");
  __syncthreads();

  int ncol = (wave << 4) + lm;
  int koff = half << 1;                        // second half-wave holds K+2,K+3
  const float* arow_lds = &ldsA[lm * APAD];

  v8f acc = {};
  #pragma unroll 8
  for (int k0 = 0; k0 < HIDDEN; k0 += 4) {
    int ka = k0 + koff;
    // A 16x4 fragment from LDS: VGPR0 = K=ka, VGPR1 = K=ka+1 (row = lm)
    v2f a = *(const v2f*)(arow_lds + ka);
    // B 4x16 fragment from global (L1/L2-resident 64KB weight matrix)
    v2f b;
    b.x = W[(size_t)ka * HIDDEN + ncol];
    b.y = W[(size_t)(ka + 1) * HIDDEN + ncol];
    acc = __builtin_amdgcn_wmma_f32_16x16x4_f32(false, a, false, b,
                                                (short)0, acc, false, false);
  }

  // C/D layout: acc[v] = C[M = v + half*8][N = lm]; uniform tail guard
  int orow0 = row0 + (half << 3);
  float* out = T + (size_t)orow0 * HIDDEN + ncol;
  if (row0 + 16 <= n) {
    #pragma unroll
    for (int v = 0; v < 8; ++v) out[(size_t)v * HIDDEN] = acc[v];
  } else {
    #pragma unroll
    for (int v = 0; v < 8; ++v)
      if (orow0 + v < n) out[(size_t)v * HIDDEN] = acc[v];
  }
}

// ---------------- aggregation: out = b + t*selfnorm + scatter(t[src]*norm) ----------------
__global__ void k_init_out(const float* __restrict__ t, const float* __restrict__ dis,
                           const float* __restrict__ bias, float* __restrict__ out, int n) {
  int idx = blockIdx.x * blockDim.x + threadIdx.x;     // one float4 per thread
  int total = n * (HIDDEN / 4);
  if (idx >= total) return;
  int node = idx >> 5;
  int c4 = (idx & 31) << 2;
  float sn = dis[node];
  sn *= sn;                                            // self-loop norm = dis^2
  float4 tv = *(const float4*)(t + (size_t)node * HIDDEN + c4);
  float4 bv = *(const float4*)(bias + c4);
  float4 r = make_float4(bv.x + tv.x * sn, bv.y + tv.y * sn,
                         bv.z + tv.z * sn, bv.w + tv.w * sn);
  *(float4*)(out + (size_t)node * HIDDEN + c4) = r;
}

// One wave per edge: 32 lanes x float4 = 128 features; L2-resident atomics.
__global__ void k_scatter(const float* __restrict__ t, const int* __restrict__ src,
                          const int* __restrict__ dst, const float* __restrict__ norm,
                          float* __restrict__ out, int e) {
  int edge = blockIdx.x * (blockDim.x >> 5) + (threadIdx.x >> 5);
  if (edge >= e) return;
  int lane = threadIdx.x & 31;
  int s = src[edge];
  int d = dst[edge];
  float w = norm[edge];
  float4 v = *(const float4*)(t + (size_t)s * HIDDEN + lane * 4);
  float* o = out + (size_t)d * HIDDEN + lane * 4;
  atomicAdd(o + 0, v.x * w);
  atomicAdd(o + 1, v.y * w);
  atomicAdd(o + 2, v.z * w);
  atomicAdd(o + 3, v.w * w);
}

// ---------------- global mean pool + linear head ----------------
__global__ void k_pool_zero(float* pooled, float* cnt) {
  int i = blockIdx.x * blockDim.x + threadIdx.x;
  if (i < NGRAPHS * HIDDEN) pooled[i] = 0.0f;
  else if (i < NGRAPHS * HIDDEN + NGRAPHS) cnt[i - NGRAPHS * HIDDEN] = 0.0f;
}

__global__ void k_pool_accum(const float* __restrict__ h, const int* __restrict__ batch,
                             float* __restrict__ pooled, float* __restrict__ cnt, int n) {
  int node = blockIdx.x * (blockDim.x >> 5) + (threadIdx.x >> 5);
  if (node >= n) return;
  int lane = threadIdx.x & 31;
  int g = batch[node];
  float4 v = *(const float4*)(h + (size_t)node * HIDDEN + lane * 4);
  float* p = pooled + (size_t)g * HIDDEN + lane * 4;
  atomicAdd(p + 0, v.x);
  atomicAdd(p + 1, v.y);
  atomicAdd(p + 2, v.z);
  atomicAdd(p + 3, v.w);
  if (lane == 0) atomicAdd(&cnt[g], 1.0f);
}

__global__ void k_final(const float* __restrict__ pooled, const float* __restrict__ cnt,
                        const float* __restrict__ Wlin, const float* __restrict__ blin,
                        float* __restrict__ out) {
  int idx = blockIdx.x * blockDim.x + threadIdx.x;
  if (idx >= NGRAPHS * NCLASSES) return;
  int g = idx / NCLASSES;
  int j = idx % NCLASSES;
  float inv = 1.0f / fmaxf(cnt[g], 1.0f);
  const float* pg = pooled + (size_t)g * HIDDEN;
  float s = blin[j];
  #pragma unroll 4
  for (int k = 0; k < HIDDEN; ++k) s += pg[k] * inv * Wlin[k * NCLASSES + j];
  out[idx] = s;
}

// ---------------- host launcher ----------------
extern "C" void kernel_launch(void* const* d_in, const int* in_sizes, int n_in,
                              void* d_out, int out_size, void* d_ws, size_t ws_size,
                              hipStream_t stream) {
  const float* x    = (const float*)d_in[0];
  const int*   edge = (const int*)d_in[1];
  const int*   batch= (const int*)d_in[2];
  const float* W1 = (const float*)d_in[3];
  const float* b1 = (const float*)d_in[4];
  const float* W2 = (const float*)d_in[5];
  const float* b2 = (const float*)d_in[6];
  const float* W3 = (const float*)d_in[7];
  const float* b3 = (const float*)d_in[8];
  const float* W4 = (const float*)d_in[9];
  const float* b4 = (const float*)d_in[10];
  const float* W5 = (const float*)d_in[11];
  const float* b5 = (const float*)d_in[12];
  const float* Wlin = (const float*)d_in[13];
  const float* blin = (const float*)d_in[14];

  const int n = in_sizes[0];          // NUM_FEAT == 1 -> x has n elements
  const int e = in_sizes[1] / 2;      // edge_index is [2, E]
  const int* src = edge;
  const int* dst = edge + e;

  // workspace carve-out (256B aligned): dis | norm | bufA | bufB | pooled | cnt
  char* ws = (char*)d_ws;
  size_t off = 0;
  auto take = [&](size_t bytes) -> char* {
    char* p = ws + off;
    off = (off + bytes + 255) & ~(size_t)255;
    return p;
  };
  float* dis    = (float*)take((size_t)n * 4);
  float* norm   = (float*)take((size_t)e * 4);
  float* bufA   = (float*)take((size_t)n * HIDDEN * 4);
  float* bufB   = (float*)take((size_t)n * HIDDEN * 4);
  float* pooled = (float*)take((size_t)NGRAPHS * HIDDEN * 4);
  float* cnt    = (float*)take((size_t)NGRAPHS * 4);
  (void)ws_size; (void)n_in; (void)out_size;

  const int T = 256;                           // 8 wave32s per block
  const int nGrid   = (n + T - 1) / T;
  const int eGrid   = (e + T - 1) / T;
  const int vecGrid = (n * (HIDDEN / 4) + T - 1) / T;
  const int edgeGrid = (e + 7) / 8;            // one edge per wave
  const int nodeWaveGrid = (n + 7) / 8;        // one node per wave
  const int mmGrid  = (n + 15) / 16;

  // symmetric GCN normalization
  k_deg_init   <<<nGrid, T, 0, stream>>>(dis, n);
  k_deg_accum  <<<eGrid, T, 0, stream>>>(dst, dis, e);
  k_deg_to_dis <<<nGrid, T, 0, stream>>>(dis, n);
  k_edge_norm  <<<eGrid, T, 0, stream>>>(src, dst, dis, norm, e);

  // layer 1 (K=1): transform then aggregate
  k_l1_transform<<<vecGrid, T, 0, stream>>>(x, W1, bufA, n);
  k_init_out    <<<vecGrid, T, 0, stream>>>(bufA, dis, b1, bufB, n);
  k_scatter     <<<edgeGrid, T, 0, stream>>>(bufA, src, dst, norm, bufB, e);

  // layers 2..5: WMMA transform + aggregate (ping-pong bufB -> bufA -> bufB)
  const float* Ws[4] = {W2, W3, W4, W5};
  const float* bs[4] = {b2, b3, b4, b5};
  for (int l = 0; l < 4; ++l) {
    k_mm128_wmma<<<mmGrid, T, 0, stream>>>(bufB, Ws[l], bufA, n);
    k_init_out  <<<vecGrid, T, 0, stream>>>(bufA, dis, bs[l], bufB, n);
    k_scatter   <<<edgeGrid, T, 0, stream>>>(bufA, src, dst, norm, bufB, e);
  }

  // global mean pool + classifier head
  const int pzGrid = (NGRAPHS * HIDDEN + NGRAPHS + T - 1) / T;
  k_pool_zero <<<pzGrid, T, 0, stream>>>(pooled, cnt);
  k_pool_accum<<<nodeWaveGrid, T, 0, stream>>>(bufB, batch, pooled, cnt, n);
  k_final     <<<(NGRAPHS * NCLASSES + T - 1) / T, T, 0, stream>>>(
      pooled, cnt, Wlin, blin, (float*)d_out);
}